// Net_48816598286344
// MI455X (gfx1250) — compile-verified
//
#include <hip/hip_runtime.h>
#include <hip/hip_bf16.h>
#include <math.h>

typedef __attribute__((ext_vector_type(2))) float v2f;
typedef __attribute__((ext_vector_type(8))) float v8f;

constexpr int NN = 100000;   // nodes
constexpr int NE = 1600000;  // edges (before self loops)
constexpr int NG = 512;      // graphs

__device__ __forceinline__ void atomAddF(float* p, float v) {
    // relaxed, device scope -> native global_atomic_add_f32 (no-return form)
    __hip_atomic_fetch_add(p, v, __ATOMIC_RELAXED, __HIP_MEMORY_SCOPE_AGENT);
}

// ---------------- degree / normalization ----------------
__global__ void k_deg_init(float* __restrict__ deg) {
    int i = blockIdx.x * blockDim.x + threadIdx.x;
    if (i < NN) deg[i] = 1.0f;  // self-loop contributes 1
}

__global__ void k_deg_accum(const int* __restrict__ dst, float* __restrict__ deg) {
    int e = blockIdx.x * blockDim.x + threadIdx.x;
    if (e < NE) atomAddF(&deg[dst[e]], 1.0f);
}

__global__ void k_deg_rsqrt(float* __restrict__ deg) {
    int i = blockIdx.x * blockDim.x + threadIdx.x;
    if (i < NN) deg[i] = rsqrtf(deg[i]);   // in-place: deg -> deg^{-1/2}
}

// ---------------- self-loop init of agg: agg[i] = disq[i]^2 * x[i] ----------------
template <int C>
__global__ void k_selfloop(const float* __restrict__ x, const float* __restrict__ disq,
                           float* __restrict__ agg) {
    int idx = blockIdx.x * blockDim.x + threadIdx.x;
    if (idx >= NN * C) return;
    int node = idx / C;
    float d = disq[node];
    agg[idx] = x[idx] * d * d;
}

// ---------------- edge scatter: agg[dst] += norm * x[src] ----------------
template <int C>
__global__ void k_edge_scatter(const float* __restrict__ x, const int* __restrict__ src,
                               const int* __restrict__ dst, const float* __restrict__ disq,
                               float* __restrict__ agg) {
    constexpr int VEC = (C >= 4) ? 4 : C;   // channels per thread
    constexpr int GPT = C / VEC;            // thread-groups per edge
    int gid = blockIdx.x * blockDim.x + threadIdx.x;
    if (gid >= NE * GPT) return;
    int e  = gid / GPT;
    int cq = (gid % GPT) * VEC;
    int s = src[e], d = dst[e];
    float nrm = disq[s] * disq[d];
    const float* xp = x   + (long long)s * C + cq;
    float*       ap = agg + (long long)d * C + cq;
    if constexpr (VEC == 4) {
        float4 xv = *reinterpret_cast<const float4*>(xp);
        atomAddF(ap + 0, xv.x * nrm);
        atomAddF(ap + 1, xv.y * nrm);
        atomAddF(ap + 2, xv.z * nrm);
        atomAddF(ap + 3, xv.w * nrm);
    } else {  // C == 2
        float2 xv = *reinterpret_cast<const float2*>(xp);
        atomAddF(ap + 0, xv.x * nrm);
        atomAddF(ap + 1, xv.y * nrm);
    }
}

// ---------------- WMMA fp32 GEMM: H = act(A[M,Cin] @ W[Cin,Cout] + b) ----------------
// One wave computes one 16x16 output tile with V_WMMA_F32_16X16X4_F32.
// A frag (16x4):  lane L -> row L%16, K-pair kb..kb+1 with kb = 2*(L/16)
// B frag (4x16):  lane L -> col L%16, K-pair kb..kb+1
// C/D (16x16):    VGPR r -> row r + 8*(L/16), col L%16
template <int Cin, int Cout, bool RELU>
__global__ void k_gemm_bias_act(const float* __restrict__ A, const float* __restrict__ W,
                                const float* __restrict__ bias, float* __restrict__ H,
                                int M) {
    const int lane = threadIdx.x & 31;
    const int wave = threadIdx.x >> 5;
    const int tile = blockIdx.x * (blockDim.x >> 5) + wave;
    constexpr int NT = Cout / 16;           // n-tiles
    const int mt = tile / NT;
    const int nt = tile % NT;
    if (mt * 16 >= M) return;               // wave-uniform guard
    const int m0 = mt * 16, n0 = nt * 16;
    const int r  = lane & 15;               // row for A, col for B
    const int kb = (lane >> 4) << 1;        // 0 or 2

    v8f acc = {};
    #pragma unroll
    for (int k = 0; k < Cin; k += 4) {
        const int k0 = k + kb, k1 = k + kb + 1;
        v2f a, b;
        a[0] = (k0 < Cin) ? A[(long long)(m0 + r) * Cin + k0] : 0.0f;
        a[1] = (k1 < Cin) ? A[(long long)(m0 + r) * Cin + k1] : 0.0f;
        b[0] = (k0 < Cin) ? W[k0 * Cout + (n0 + r)] : 0.0f;
        b[1] = (k1 < Cin) ? W[k1 * Cout + (n0 + r)] : 0.0f;
        acc = __builtin_amdgcn_wmma_f32_16x16x4_f32(false, a, false, b,
                                                    (short)0, acc, false, false);
    }

    const int row_hi = (lane >> 4) << 3;    // 0 or 8
    const int col = n0 + (lane & 15);
    const float bc = bias[col];
    #pragma unroll
    for (int rr = 0; rr < 8; ++rr) {
        float v = acc[rr] + bc;
        if constexpr (RELU) v = fmaxf(v, 0.0f);
        H[(long long)(m0 + rr + row_hi) * Cout + col] = v;
    }
}

// ---------------- segment_max pooling over sorted batch ----------------
__global__ void k_pool_max(const float* __restrict__ H,      // [NN,128]
                           const int* __restrict__ batch,    // [NN], sorted
                           float* __restrict__ pooled) {     // [NG,128]
    const int g = blockIdx.x;
    const int c = threadIdx.x;   // 128 threads
    __shared__ int slo, shi;
    if (threadIdx.x == 0) {
        int lo = 0, hi = NN;
        while (lo < hi) { int mid = (lo + hi) >> 1; if (batch[mid] < g) lo = mid + 1; else hi = mid; }
        slo = lo;
        int lo2 = lo, hi2 = NN;
        while (lo2 < hi2) { int mid = (lo2 + hi2) >> 1; if (batch[mid] < g + 1) lo2 = mid + 1; else hi2 = mid; }
        shi = lo2;
    }
    __syncthreads();
    float m = -INFINITY;
    for (int i = slo; i < shi; ++i)
        m = fmaxf(m, H[(long long)i * 128 + c]);
    pooled[g * 128 + c] = isfinite(m) ? m : 0.0f;   // empty segment -> 0
}

// ---------------- tiny final layer: out = fc5 @ W6 + b6 ----------------
__global__ void k_fc6(const float* __restrict__ A, const float* __restrict__ W,
                      const float* __restrict__ b, float* __restrict__ out) {
    int idx = blockIdx.x * blockDim.x + threadIdx.x;
    if (idx >= NG * 10) return;
    int g = idx / 10, c = idx % 10;
    float acc = b[c];
    #pragma unroll
    for (int k = 0; k < 64; ++k) acc += A[g * 64 + k] * W[k * 10 + c];
    out[idx] = acc;
}

// ---------------- launch ----------------
static inline int cdiv(long long a, int b) { return (int)((a + b - 1) / b); }

extern "C" void kernel_launch(void* const* d_in, const int* in_sizes, int n_in,
                              void* d_out, int out_size, void* d_ws, size_t ws_size,
                              hipStream_t stream) {
    const float* x     = (const float*)d_in[0];
    const int*   ei    = (const int*)d_in[1];
    const int*   batch = (const int*)d_in[2];
    const float* W1 = (const float*)d_in[3],  *b1 = (const float*)d_in[4];
    const float* W2 = (const float*)d_in[5],  *b2 = (const float*)d_in[6];
    const float* W3 = (const float*)d_in[7],  *b3 = (const float*)d_in[8];
    const float* W4 = (const float*)d_in[9],  *b4 = (const float*)d_in[10];
    const float* W5 = (const float*)d_in[11], *b5 = (const float*)d_in[12];
    const float* W6 = (const float*)d_in[13], *b6 = (const float*)d_in[14];
    const int* src = ei;             // edge_index[0]
    const int* dst = ei + NE;        // edge_index[1]
    float* out = (float*)d_out;

    // workspace carve (floats)
    float* wsf    = (float*)d_ws;
    float* disq   = wsf;                            // NN
    float* agg    = disq + NN;                      // NN*64 (max Cin)
    float* bufA   = agg  + (size_t)NN * 64;         // NN*64  (H1,H3)
    float* bufB   = bufA + (size_t)NN * 64;         // NN*128 (H2,H4)
    float* pooled = bufB + (size_t)NN * 128;        // NG*128
    float* fc5    = pooled + (size_t)NG * 128;      // NG*64

    const int TB = 256;

    // degrees -> D^{-1/2}
    k_deg_init <<<cdiv(NN, TB), TB, 0, stream>>>(disq);
    k_deg_accum<<<cdiv(NE, TB), TB, 0, stream>>>(dst, disq);
    k_deg_rsqrt<<<cdiv(NN, TB), TB, 0, stream>>>(disq);

    auto gemm_blocks = [&](int M, int Cout) { return cdiv((long long)(M / 16) * (Cout / 16), 8); };

    // layer 1: [N,2] -> [N,16] (bufA)
    k_selfloop<2>      <<<cdiv((long long)NN * 2, TB), TB, 0, stream>>>(x, disq, agg);
    k_edge_scatter<2>  <<<cdiv((long long)NE * 1, TB), TB, 0, stream>>>(x, src, dst, disq, agg);
    k_gemm_bias_act<2, 16, true><<<gemm_blocks(NN, 16), TB, 0, stream>>>(agg, W1, b1, bufA, NN);

    // layer 2: [N,16] -> [N,32] (bufB)
    k_selfloop<16>     <<<cdiv((long long)NN * 16, TB), TB, 0, stream>>>(bufA, disq, agg);
    k_edge_scatter<16> <<<cdiv((long long)NE * 4, TB), TB, 0, stream>>>(bufA, src, dst, disq, agg);
    k_gemm_bias_act<16, 32, true><<<gemm_blocks(NN, 32), TB, 0, stream>>>(agg, W2, b2, bufB, NN);

    // layer 3: [N,32] -> [N,64] (bufA)
    k_selfloop<32>     <<<cdiv((long long)NN * 32, TB), TB, 0, stream>>>(bufB, disq, agg);
    k_edge_scatter<32> <<<cdiv((long long)NE * 8, TB), TB, 0, stream>>>(bufB, src, dst, disq, agg);
    k_gemm_bias_act<32, 64, true><<<gemm_blocks(NN, 64), TB, 0, stream>>>(agg, W3, b3, bufA, NN);

    // layer 4: [N,64] -> [N,128] (bufB)
    k_selfloop<64>     <<<cdiv((long long)NN * 64, TB), TB, 0, stream>>>(bufA, disq, agg);
    k_edge_scatter<64> <<<cdiv((long long)NE * 16, TB), TB, 0, stream>>>(bufA, src, dst, disq, agg);
    k_gemm_bias_act<64, 128, true><<<gemm_blocks(NN, 128), TB, 0, stream>>>(agg, W4, b4, bufB, NN);

    // segment_max pooling -> [512,128]
    k_pool_max<<<NG, 128, 0, stream>>>(bufB, batch, pooled);

    // head: fc5 (WMMA) + fc6 (scalar, Cout=10)
    k_gemm_bias_act<128, 64, true><<<gemm_blocks(NG, 64), TB, 0, stream>>>(pooled, W5, b5, fc5, NG);
    k_fc6<<<cdiv(NG * 10, TB), TB, 0, stream>>>(fc5, W6, b6, out);
}